// VectorsFromMask_83760452206670
// MI455X (gfx1250) — compile-verified
//
#include <hip/hip_runtime.h>

#ifndef __has_builtin
#define __has_builtin(x) 0
#endif

// Problem geometry (fixed by the reference).
#define B_      8
#define C_      128
#define M_      16
#define S_      16384          // H*W = 128*128
#define MOUT    15             // mask channels 1..15

// Pipeline geometry.
#define NTHREADS   256                       // 8 wave32s
#define CHUNK      2048                      // floats per LDS buffer
#define NCHUNKS    (S_ / CHUNK)              // 8
#define VEC_ITERS  (CHUNK / (NTHREADS * 4))  // 2 x B128 per stream per chunk
#define OPS_PER_CHUNK (VEC_ITERS * 2)        // 4 async DMA ops / thread / chunk

#if __has_builtin(__builtin_amdgcn_global_load_async_to_lds_b128)
#define USE_ASYNC 1
#else
#define USE_ASYNC 0
#endif

// Async-to-LDS builtins take (v4i addrspace(1)*, v4i addrspace(3)*, imm, imm).
typedef int v4i __attribute__((ext_vector_type(4)));
typedef __attribute__((address_space(1))) v4i g_v4i;
typedef __attribute__((address_space(3))) v4i l_v4i;

__device__ __forceinline__ void copy16_to_lds(const void* g, void* l) {
#if USE_ASYNC
  // gfx1250 async DMA: global -> LDS, 16 bytes/lane, tracked by ASYNCcnt.
  __builtin_amdgcn_global_load_async_to_lds_b128((g_v4i*)g, (l_v4i*)l,
                                                 /*offset=*/0, /*cpol=*/0);
#else
  *(float4*)l = *(const float4*)g;  // synchronous fallback
#endif
}

__device__ __forceinline__ void wait_async_keep_next() {
#if USE_ASYNC
#if __has_builtin(__builtin_amdgcn_s_wait_asynccnt)
  __builtin_amdgcn_s_wait_asynccnt(OPS_PER_CHUNK);  // current chunk done; next in flight
#else
  asm volatile("s_wait_asynccnt 4" ::: "memory");
#endif
#endif
}

__device__ __forceinline__ void wait_async_all() {
#if USE_ASYNC
#if __has_builtin(__builtin_amdgcn_s_wait_asynccnt)
  __builtin_amdgcn_s_wait_asynccnt(0);
#else
  asm volatile("s_wait_asynccnt 0" ::: "memory");
#endif
#endif
}

// ---------------------------------------------------------------------------
// Main pass: one block per (b, c). Stream the 16384-float row through a
// double-buffered LDS pipeline fed by async-to-LDS DMA; 15 running maxima
// per thread; wave shfl_xor max tree + cross-wave LDS reduction.
// (Defined first so the disassembly snippet shows this kernel.)
// ---------------------------------------------------------------------------
__global__ void __launch_bounds__(NTHREADS)
region_max_kernel(const float* __restrict__ enc,
                  const unsigned* __restrict__ pmask,
                  float* __restrict__ out) {
  const int c   = blockIdx.x;
  const int b   = blockIdx.y;
  const int tid = threadIdx.x;

  __shared__ float    s_enc[2][CHUNK];
  __shared__ unsigned s_msk[2][CHUNK];
  __shared__ float    s_red[NTHREADS / 32][MOUT];

  const float*    genc = enc   + ((size_t)b * C_ + c) * S_;
  const unsigned* gmsk = pmask + (size_t)b * S_;

  const float NEG = -__builtin_inff();
  float acc[MOUT];
#pragma unroll
  for (int m = 0; m < MOUT; ++m) acc[m] = NEG;

  // Prefetch chunk 0 into buffer 0.
#pragma unroll
  for (int j = 0; j < VEC_ITERS; ++j) {
    const int idx = (j * NTHREADS + tid) * 4;
    copy16_to_lds(genc + idx, &s_enc[0][idx]);
    copy16_to_lds(gmsk + idx, &s_msk[0][idx]);
  }

  for (int k = 0; k < NCHUNKS; ++k) {
    const int cur = k & 1;
    if (k + 1 < NCHUNKS) {
      const int base = (k + 1) * CHUNK;
#pragma unroll
      for (int j = 0; j < VEC_ITERS; ++j) {
        const int idx = (j * NTHREADS + tid) * 4;
        copy16_to_lds(genc + base + idx, &s_enc[cur ^ 1][idx]);
        copy16_to_lds(gmsk + base + idx, &s_msk[cur ^ 1][idx]);
      }
      wait_async_keep_next();   // retire current chunk's 4 DMA ops (in-order)
    } else {
      wait_async_all();
    }
    __syncthreads();            // all waves' DMA for `cur` complete

    // Vectorized LDS reads: 4 consecutive positions per b128 load.
#pragma unroll
    for (int j = 0; j < VEC_ITERS; ++j) {
      const int idx = (j * NTHREADS + tid) * 4;
      const float4 e4 = *(const float4*)&s_enc[cur][idx];
      const uint4  w4 = *(const uint4*)&s_msk[cur][idx];
      const float    e[4] = {e4.x, e4.y, e4.z, e4.w};
      const unsigned w[4] = {w4.x, w4.y, w4.z, w4.w};
#pragma unroll
      for (int i = 0; i < 4; ++i) {
#pragma unroll
        for (int m = 0; m < MOUT; ++m) {
          const float v = ((w[i] >> m) & 1u) ? e[i] : NEG;  // cndmask
          acc[m] = fmaxf(acc[m], v);                        // v_max_num_f32
        }
      }
    }
    __syncthreads();            // buffer `cur` free for next prefetch round
  }

  // Wave-level max tree (wave32).
#pragma unroll
  for (int m = 0; m < MOUT; ++m) {
    float v = acc[m];
#pragma unroll
    for (int off = 16; off > 0; off >>= 1)
      v = fmaxf(v, __shfl_xor(v, off, 32));
    acc[m] = v;
  }

  const int wave = tid >> 5;
  const int lane = tid & 31;
  if (lane == 0) {
#pragma unroll
    for (int m = 0; m < MOUT; ++m) s_red[wave][m] = acc[m];
  }
  __syncthreads();

  if (tid < MOUT) {
    float v = s_red[0][tid];
#pragma unroll
    for (int w = 1; w < NTHREADS / 32; ++w) v = fmaxf(v, s_red[w][tid]);
    // out layout: [B, C, MOUT, 1] flat
    out[((size_t)b * C_ + c) * MOUT + tid] = v;
  }
}

// ---------------------------------------------------------------------------
// Pre-pass: pack 15 int32 0/1 masks per (b, position) into one 15-bit word.
// 8 MB of int32 masks -> 512 KB of packed words in d_ws (L2-resident after).
// ---------------------------------------------------------------------------
__global__ void pack_masks_kernel(const int* __restrict__ masks,
                                  unsigned* __restrict__ packed) {
  const int t = blockIdx.x * blockDim.x + threadIdx.x;  // 0 .. B_*S_-1
  if (t >= B_ * S_) return;
  const int b = t >> 14;          // t / S_
  const int p = t & (S_ - 1);     // t % S_
  const int* mp = masks + (size_t)b * M_ * S_ + p;
  unsigned w = 0;
#pragma unroll
  for (int m = 1; m < M_; ++m)
    w |= (mp[(size_t)m * S_] != 0 ? 1u : 0u) << (m - 1);
  packed[t] = w;
}

extern "C" void kernel_launch(void* const* d_in, const int* in_sizes, int n_in,
                              void* d_out, int out_size, void* d_ws, size_t ws_size,
                              hipStream_t stream) {
  const float* enc    = (const float*)d_in[0];  // [8,128,128,128] f32
  const int*   masks  = (const int*)d_in[1];    // [8,16,128,128]  i32
  float*       out    = (float*)d_out;          // [8,128,15,1]    f32
  unsigned*    packed = (unsigned*)d_ws;        // [8,16384] u32 (512 KB)

  pack_masks_kernel<<<(B_ * S_ + 255) / 256, 256, 0, stream>>>(masks, packed);

  dim3 grid(C_, B_);
  region_max_kernel<<<grid, NTHREADS, 0, stream>>>(enc, packed, out);
}